// GCN_71287867179163
// MI455X (gfx1250) — compile-verified
//
#include <hip/hip_runtime.h>
#include <hip/hip_bf16.h>

// ---------------------------------------------------------------------------
// GCN forward on MI455X (gfx1250, wave32).
//   h1 = relu(Prop(x @ W1) + b1); h2 = relu(Prop(h1 @ W2) + b2)
//   out = log_softmax(mean_pool(h2) @ Wlin + blin)
// Dense transforms: V_WMMA_F32_16X16X4_F32 (fp32 tensor path).
// Sparse propagation: float4-vectorized gathers (global_load_b128) +
// L2-resident global_atomic_add_f32 scatters.
// ---------------------------------------------------------------------------

typedef float v8f __attribute__((ext_vector_type(8)));
typedef float v2f __attribute__((ext_vector_type(2)));

#define F_IN 128
#define HDIM 64
#define HQ   16      // HDIM / 4 (float4 chunks per row)
#define NCLS 10

// ---------------- degree / normalization ----------------
__global__ void k_init_deg(float* deg, int n) {
    int i = blockIdx.x * blockDim.x + threadIdx.x;
    if (i < n) deg[i] = 1.0f;                       // self-loop contribution
}

__global__ void k_count_deg(const int* __restrict__ dst, float* deg, int e) {
    int i = blockIdx.x * blockDim.x + threadIdx.x;
    if (i < e) atomicAdd(&deg[dst[i]], 1.0f);
}

__global__ void k_make_dinv(float* deg, int n) {
    int i = blockIdx.x * blockDim.x + threadIdx.x;
    if (i < n) {
        float d = deg[i];
        deg[i] = (d > 0.0f) ? __frsqrt_rn(d) : 0.0f; // in place: deg -> dinv
    }
}

// ---------------- dense transform: Y[N,64] = X[N,K] @ W[K,64] --------------
// One wave computes a 16x64 tile with 4 v8f accumulators.
// A fragment (16x4 f32): lane&15 = M; lane>>4 selects K pair {0,1} vs {2,3}.
// B fragment (4x16): lane&15 = N; lane>>4 selects K row pair.
// C/D: VGPR r holds M = r + 8*(lane>>4), N = lane&15.
__global__ void __launch_bounds__(256) k_gemm_wmma(const float* __restrict__ X,
                                                   const float* __restrict__ W,
                                                   float* __restrict__ Y,
                                                   int n, int K) {
    int gtid    = blockIdx.x * blockDim.x + threadIdx.x;
    int wave    = gtid >> 5;              // global wave id = row-block id
    int lane    = threadIdx.x & 31;
    int rowBase = wave * 16;
    if (rowBase >= n) return;             // wave-uniform exit: EXEC stays all-1s

    int m  = lane & 15;                   // M (A) and N (B/C/D) index
    int kk = (lane >> 4) << 1;            // 0 or 2: K-pair select

    v8f c0 = {}, c1 = {}, c2 = {}, c3 = {};

    for (int k = 0; k < K; k += 4) {
        v2f a;
        const float* xp = X + (size_t)(rowBase + m) * K + (k + kk);
        a.x = xp[0];
        a.y = xp[1];

        const float* wp = W + (size_t)(k + kk) * HDIM + m;
        v2f b0, b1, b2, b3;
        b0.x = wp[0];   b0.y = wp[HDIM + 0];
        b1.x = wp[16];  b1.y = wp[HDIM + 16];
        b2.x = wp[32];  b2.y = wp[HDIM + 32];
        b3.x = wp[48];  b3.y = wp[HDIM + 48];

        c0 = __builtin_amdgcn_wmma_f32_16x16x4_f32(false, a, false, b0, (short)0, c0, false, false);
        c1 = __builtin_amdgcn_wmma_f32_16x16x4_f32(false, a, false, b1, (short)0, c1, false, false);
        c2 = __builtin_amdgcn_wmma_f32_16x16x4_f32(false, a, false, b2, (short)0, c2, false, false);
        c3 = __builtin_amdgcn_wmma_f32_16x16x4_f32(false, a, false, b3, (short)0, c3, false, false);
    }

    int rOff = (lane >> 4) << 3;          // upper half-wave holds M = r + 8
    for (int r = 0; r < 8; ++r) {
        size_t row = (size_t)(rowBase + rOff + r);
        Y[row * HDIM +  0 + m] = c0[r];
        Y[row * HDIM + 16 + m] = c1[r];
        Y[row * HDIM + 32 + m] = c2[r];
        Y[row * HDIM + 48 + m] = c3[r];
    }
}

// ---------------- propagation: a = bias + selfloop + scatter(edges) --------
// float4 per thread: feature row handled as 16 float4 chunks.
__global__ void k_self_bias(float4* __restrict__ a, const float4* __restrict__ t,
                            const float* __restrict__ dinv,
                            const float4* __restrict__ bias, int n) {
    int tid = blockIdx.x * blockDim.x + threadIdx.x;
    if (tid < n * HQ) {
        int i = tid >> 4;
        int q = tid & 15;
        float di = dinv[i];
        float w  = di * di;                    // self-loop norm = dinv[i]^2
        float4 v = t[tid];
        float4 b = bias[q];
        float4 r;
        r.x = w * v.x + b.x;
        r.y = w * v.y + b.y;
        r.z = w * v.z + b.z;
        r.w = w * v.w + b.w;
        a[tid] = r;
    }
}

__global__ void k_scatter(float* __restrict__ a, const float4* __restrict__ t,
                          const float* __restrict__ dinv,
                          const int* __restrict__ src,
                          const int* __restrict__ dst, int e) {
    long long tid = (long long)blockIdx.x * blockDim.x + threadIdx.x;
    if (tid >= (long long)e * HQ) return;
    int ei = (int)(tid >> 4);
    int q  = (int)(tid & 15);
    int s = src[ei];
    int d = dst[ei];
    float w = dinv[s] * dinv[d];
    float4 v = t[(size_t)s * HQ + q];          // global_load_b128 gather
    float* p = a + (size_t)d * HDIM + q * 4;
    atomicAdd(p + 0, w * v.x);
    atomicAdd(p + 1, w * v.y);
    atomicAdd(p + 2, w * v.z);
    atomicAdd(p + 3, w * v.w);
}

__global__ void k_relu(float4* a, int n4) {
    int tid = blockIdx.x * blockDim.x + threadIdx.x;
    if (tid < n4) {
        float4 v = a[tid];
        v.x = fmaxf(v.x, 0.0f);
        v.y = fmaxf(v.y, 0.0f);
        v.z = fmaxf(v.z, 0.0f);
        v.w = fmaxf(v.w, 0.0f);
        a[tid] = v;
    }
}

// ---------------- pooling ----------------
__global__ void k_zero(float* p, int n) {
    int tid = blockIdx.x * blockDim.x + threadIdx.x;
    if (tid < n) p[tid] = 0.0f;
}

__global__ void k_pool_sum(const float4* __restrict__ a,
                           const int* __restrict__ batch,
                           float* __restrict__ pooled, int n) {
    int tid = blockIdx.x * blockDim.x + threadIdx.x;
    if (tid < n * HQ) {
        int i = tid >> 4;
        int q = tid & 15;
        float4 v = a[tid];
        float* p = pooled + (size_t)batch[i] * HDIM + q * 4;
        atomicAdd(p + 0, v.x);
        atomicAdd(p + 1, v.y);
        atomicAdd(p + 2, v.z);
        atomicAdd(p + 3, v.w);
    }
}

__global__ void k_pool_cnt(const int* __restrict__ batch, float* cnt, int n) {
    int i = blockIdx.x * blockDim.x + threadIdx.x;
    if (i < n) atomicAdd(&cnt[batch[i]], 1.0f);
}

// ---------------- head: mean, linear, log_softmax ----------------
__global__ void k_head(const float* __restrict__ pooled,
                       const float* __restrict__ cnt,
                       const float* __restrict__ Wlin,
                       const float* __restrict__ blin,
                       float* __restrict__ out, int g) {
    int gi = blockIdx.x * blockDim.x + threadIdx.x;
    if (gi >= g) return;
    float inv = 1.0f / fmaxf(cnt[gi], 1.0f);
    float logit[NCLS];
#pragma unroll
    for (int c = 0; c < NCLS; ++c) logit[c] = blin[c];
    for (int f = 0; f < HDIM; ++f) {
        float v = pooled[(size_t)gi * HDIM + f] * inv;
#pragma unroll
        for (int c = 0; c < NCLS; ++c) logit[c] += v * Wlin[f * NCLS + c];
    }
    float mx = logit[0];
#pragma unroll
    for (int c = 1; c < NCLS; ++c) mx = fmaxf(mx, logit[c]);
    float s = 0.0f;
#pragma unroll
    for (int c = 0; c < NCLS; ++c) s += __expf(logit[c] - mx);
    float lse = mx + __logf(s);
#pragma unroll
    for (int c = 0; c < NCLS; ++c) out[(size_t)gi * NCLS + c] = logit[c] - lse;
}

// ---------------------------------------------------------------------------
extern "C" void kernel_launch(void* const* d_in, const int* in_sizes, int n_in,
                              void* d_out, int out_size, void* d_ws, size_t ws_size,
                              hipStream_t stream) {
    const float* x     = (const float*)d_in[0];
    const int*   ei    = (const int*)d_in[1];     // [2, E] flat: row0=src, row1=dst
    const int*   batch = (const int*)d_in[2];
    const float* W1    = (const float*)d_in[3];
    const float* b1    = (const float*)d_in[4];
    const float* W2    = (const float*)d_in[5];
    const float* b2    = (const float*)d_in[6];
    const float* Wlin  = (const float*)d_in[7];
    const float* blin  = (const float*)d_in[8];
    float*       out   = (float*)d_out;

    const int N = in_sizes[0] / F_IN;
    const int E = in_sizes[1] / 2;
    const int G = out_size / NCLS;

    const int* src = ei;
    const int* dst = ei + E;

    // workspace carve-out (256B aligned)
    char* ws = (char*)d_ws;
    size_t off = 0;
    auto carve = [&](size_t bytes) -> float* {
        float* p = (float*)(ws + off);
        off = (off + bytes + 255) & ~(size_t)255;
        return p;
    };
    float* dinv   = carve((size_t)N * sizeof(float));          // deg -> dinv in place
    float* tbuf   = carve((size_t)N * HDIM * sizeof(float));   // X @ W
    float* abuf   = carve((size_t)N * HDIM * sizeof(float));   // aggregated
    float* pooled = carve((size_t)G * HDIM * sizeof(float));
    float* cnt    = carve((size_t)G * sizeof(float));
    (void)ws_size;

    const int TB = 256;
    int gN   = (N + TB - 1) / TB;
    int gE   = (E + TB - 1) / TB;
    int gNQ  = (N * HQ + TB - 1) / TB;                          // float4 over [N,64]
    long long eq = (long long)E * HQ;
    int gEQ  = (int)((eq + TB - 1) / TB);                       // float4 over [E,64]
    int waves = (N + 15) / 16;
    int gW   = (waves * 32 + TB - 1) / TB;

    // --- degree / symmetric normalization ---
    k_init_deg<<<gN, TB, 0, stream>>>(dinv, N);
    k_count_deg<<<gE, TB, 0, stream>>>(dst, dinv, E);
    k_make_dinv<<<gN, TB, 0, stream>>>(dinv, N);

    // --- layer 1 ---
    k_gemm_wmma<<<gW, TB, 0, stream>>>(x, W1, tbuf, N, F_IN);
    k_self_bias<<<gNQ, TB, 0, stream>>>((float4*)abuf, (const float4*)tbuf, dinv,
                                        (const float4*)b1, N);
    k_scatter<<<gEQ, TB, 0, stream>>>(abuf, (const float4*)tbuf, dinv, src, dst, E);
    k_relu<<<gNQ, TB, 0, stream>>>((float4*)abuf, N * HQ);

    // --- layer 2 (reuse tbuf/abuf) ---
    k_gemm_wmma<<<gW, TB, 0, stream>>>(abuf, W2, tbuf, N, HDIM);
    k_self_bias<<<gNQ, TB, 0, stream>>>((float4*)abuf, (const float4*)tbuf, dinv,
                                        (const float4*)b2, N);
    k_scatter<<<gEQ, TB, 0, stream>>>(abuf, (const float4*)tbuf, dinv, src, dst, E);
    k_relu<<<gNQ, TB, 0, stream>>>((float4*)abuf, N * HQ);

    // --- global mean pool + head ---
    k_zero<<<(G * HDIM + TB - 1) / TB, TB, 0, stream>>>(pooled, G * HDIM);
    k_zero<<<(G + TB - 1) / TB, TB, 0, stream>>>(cnt, G);
    k_pool_sum<<<gNQ, TB, 0, stream>>>((const float4*)abuf, batch, pooled, N);
    k_pool_cnt<<<gN, TB, 0, stream>>>(batch, cnt, N);
    k_head<<<(G + TB - 1) / TB, TB, 0, stream>>>(pooled, cnt, Wlin, blin, out, G);
}